// EdgeCut_28346784153686
// MI455X (gfx1250) — compile-verified
//
#include <hip/hip_runtime.h>
#include <hip/hip_bf16.h>
#include <math.h>

// ---------------- problem constants (fixed by the reference) ----------------
constexpr int    NV  = 3072;               // nodes
constexpr int    FIN = 512;                // input features
constexpr int    H   = 256;                // hidden
constexpr int    HF  = 64;                 // edge-fc hidden
constexpr int    OD  = 16;                 // output classes
constexpr size_t NN  = (size_t)NV * NV;    // 9,437,184
constexpr int    KB  = 64;                 // k-chunk for LDS-staged GEMM

typedef float v2f __attribute__((ext_vector_type(2)));
typedef float v8f __attribute__((ext_vector_type(8)));
typedef int   v4i __attribute__((vector_size(16)));

enum { GF_RELU = 1, GF_ELU = 2, GF_ACC = 4 };

#if defined(__has_builtin)
#if __has_builtin(__builtin_amdgcn_global_load_async_to_lds_b128)
#define HAVE_ASYNC_LDS 1
#endif
#endif
#ifndef HAVE_ASYNC_LDS
#define HAVE_ASYNC_LDS 0
#endif

#if HAVE_ASYNC_LDS
typedef __attribute__((address_space(1))) v4i* v4i_gptr;   // global int4*
typedef __attribute__((address_space(3))) v4i* v4i_lptr;   // LDS int4*
#endif

__device__ __forceinline__ void wait_async_lds() {
#if HAVE_ASYNC_LDS
  asm volatile("s_wait_asynccnt 0x0" ::: "memory");
#endif
}

struct Scal {
  int npos; int klo; int khi; float frac;
  unsigned pref_lo; unsigned pref_hi;
  int rem_lo; int rem_hi; float cut;
};

// ---------------- fp32 WMMA GEMM (direct loads): C = act(A@B [+C] + bias) ---
// A: MxK row-major (optionally row-scaled), B: KxN row-major, C: MxN row-major.
template <int NT>
__global__ __launch_bounds__(256) void gemm_wmma(
    const float* __restrict__ A, int lda, const float* __restrict__ B, int ldb,
    float* __restrict__ C, int ldc, int M, int K, int Nc,
    const float* __restrict__ bias, const float* __restrict__ rowscale, int flags)
{
  const int lane    = threadIdx.x & 31;
  const int wid     = blockIdx.x * (blockDim.x >> 5) + (threadIdx.x >> 5);
  const int groupsN = Nc / (16 * NT);
  const int tilesM  = M / 16;
  if (wid >= tilesM * groupsN) return;
  const int ti   = wid / groupsN;
  const int tg   = wid % groupsN;
  const int row0 = ti * 16;
  const int col0 = tg * (16 * NT);
  const int half = lane >> 4;      // 0: lanes 0-15, 1: lanes 16-31
  const int r    = lane & 15;

  v8f acc[NT];
#pragma unroll
  for (int t = 0; t < NT; ++t) {
    if (flags & GF_ACC) {
#pragma unroll
      for (int v = 0; v < 8; ++v)
        acc[t][v] = C[(size_t)(row0 + v + 8 * half) * ldc + (col0 + t * 16 + r)];
    } else {
      v8f z = {};
      acc[t] = z;
    }
  }

  const float rs    = rowscale ? rowscale[row0 + r] : 1.0f;
  const float* Arow = A + (size_t)(row0 + r) * lda;

  for (int kk = 0; kk < K; kk += 4) {
    const int ka = kk + half * 2;           // A 16x4 layout: v0 = {K0|K2}, v1 = {K1|K3}
    v2f a = *(const v2f*)(Arow + ka);       // 8B-aligned vector load
    a = a * rs;
    const float* Bp = B + (size_t)ka * ldb + col0 + r;
#pragma unroll
    for (int t = 0; t < NT; ++t) {
      v2f b;
      b.x = Bp[t * 16];
      b.y = Bp[t * 16 + ldb];
      acc[t] = __builtin_amdgcn_wmma_f32_16x16x4_f32(
          false, a, false, b, (short)0, acc[t], false, false);
    }
  }

#pragma unroll
  for (int t = 0; t < NT; ++t) {
    const int col = col0 + t * 16 + r;
    const float bv = bias ? bias[col] : 0.0f;
#pragma unroll
    for (int v = 0; v < 8; ++v) {
      const int row = row0 + v + 8 * half;
      float val = acc[t][v] + bv;
      if (flags & GF_RELU) val = fmaxf(val, 0.0f);
      if (flags & GF_ELU)  val = (val > 0.0f) ? val : expm1f(val);
      C[(size_t)row * ldc + col] = val;
    }
  }
}

// ---------------- fp32 WMMA GEMM, K = NV, B staged in LDS asynchronously ----
// Block (8 waves) computes a 128 x 64 tile; B slab (KB x 64) is double-buffered
// in LDS via GLOBAL_LOAD_ASYNC_TO_LDS_B128 (ASYNCcnt + barrier sync), so the
// next chunk's loads overlap this chunk's WMMA work and B is served from LDS.
__global__ __launch_bounds__(256) void gemm_nn_async(
    const float* __restrict__ A,           // NV x NV, lda = NV
    const float* __restrict__ B,           // NV x Nc, ldb = Nc
    float* __restrict__ C, int Nc,         // ldc = Nc
    const float* __restrict__ bias, const float* __restrict__ rowscale, int flags)
{
  __shared__ float lb[2][KB * 64];         // 32 KB
  const int tid  = threadIdx.x;
  const int lane = tid & 31;
  const int wv   = tid >> 5;               // 0..7
  const int nb   = Nc / 64;
  const int bcol = blockIdx.x % nb;
  const int brow = blockIdx.x / nb;
  const int row0 = brow * 128 + wv * 16;
  const int col0 = bcol * 64;
  const int half = lane >> 4;
  const int r    = lane & 15;

  auto issue = [&](int buf, int kk0) {
#pragma unroll
    for (int c = 0; c < 4; ++c) {          // 256 thr x 4 x 16B = 16 KB slab
      const int f4   = tid + c * 256;
      const int krow = f4 >> 4;            // 0..63
      const int col4 = (f4 & 15) << 2;     // 0..60
      const float* g = B + (size_t)(kk0 + krow) * Nc + col0 + col4;
      float* l = &lb[buf][krow * 64 + col4];
#if HAVE_ASYNC_LDS
      __builtin_amdgcn_global_load_async_to_lds_b128(
          (v4i_gptr)g, (v4i_lptr)l, 0, 0);
#else
      *(float4*)l = *(const float4*)g;
#endif
    }
  };

  v8f acc[4];
#pragma unroll
  for (int t = 0; t < 4; ++t) {
    if (flags & GF_ACC) {
#pragma unroll
      for (int v = 0; v < 8; ++v)
        acc[t][v] = C[(size_t)(row0 + v + 8 * half) * Nc + (col0 + t * 16 + r)];
    } else {
      v8f z = {};
      acc[t] = z;
    }
  }

  const float rs    = rowscale ? rowscale[row0 + r] : 1.0f;
  const float* Arow = A + (size_t)(row0 + r) * NV;

  constexpr int NCH = NV / KB;             // 48 chunks
  issue(0, 0);
  wait_async_lds();
  __syncthreads();

  for (int ch = 0; ch < NCH; ++ch) {
    const int buf = ch & 1;
    if (ch + 1 < NCH) issue(buf ^ 1, (ch + 1) * KB);

    const float* Ab = Arow + ch * KB;
    const float* Lb = &lb[buf][0];
#pragma unroll 4
    for (int kk = 0; kk < KB; kk += 4) {
      const int ka = kk + half * 2;
      v2f a = *(const v2f*)(Ab + ka);
      a = a * rs;
      const float* Bp = Lb + ka * 64 + r;
#pragma unroll
      for (int t = 0; t < 4; ++t) {
        v2f b;
        b.x = Bp[t * 16];
        b.y = Bp[t * 16 + 64];
        acc[t] = __builtin_amdgcn_wmma_f32_16x16x4_f32(
            false, a, false, b, (short)0, acc[t], false, false);
      }
    }
    wait_async_lds();                      // own async loads for next buf done
    __syncthreads();                       // everyone's loads visible
  }

#pragma unroll
  for (int t = 0; t < 4; ++t) {
    const int col = col0 + t * 16 + r;
    const float bv = bias ? bias[col] : 0.0f;
#pragma unroll
    for (int v = 0; v < 8; ++v) {
      const int row = row0 + v + 8 * half;
      float val = acc[t][v] + bv;
      if (flags & GF_RELU) val = fmaxf(val, 0.0f);
      if (flags & GF_ELU)  val = (val > 0.0f) ? val : expm1f(val);
      C[(size_t)row * Nc + col] = val;
    }
  }
}

// ---------------- elementwise / reduction kernels ---------------------------
__global__ void fill_f_k(float* __restrict__ p, float v, size_t n) {
  size_t i = (size_t)blockIdx.x * blockDim.x + threadIdx.x;
  if (i < n) p[i] = v;
}
__global__ void fill_u_k(unsigned* __restrict__ p, size_t n) {
  size_t i = (size_t)blockIdx.x * blockDim.x + threadIdx.x;
  if (i < n) p[i] = 0u;
}

__global__ __launch_bounds__(256) void l2norm_k(const float* __restrict__ src,
                                                const float* __restrict__ res,
                                                float* __restrict__ dst) {
  __shared__ float sh[256];
  const int i = blockIdx.x, t = threadIdx.x;
  float v = src[(size_t)i * H + t];
  if (res) v += res[(size_t)i * H + t];
  sh[t] = v * v;
  __syncthreads();
  for (int s = 128; s > 0; s >>= 1) { if (t < s) sh[t] += sh[t + s]; __syncthreads(); }
  const float nrm = fmaxf(sqrtf(sh[0]), 1e-12f);
  dst[(size_t)i * H + t] = v / nrm;
}

__global__ void edge_s_k(const float* __restrict__ hfc, const float* __restrict__ a,
                         float* __restrict__ s1, float* __restrict__ s2) {
  int i = blockIdx.x * blockDim.x + threadIdx.x;
  if (i >= NV) return;
  const float* hr = hfc + (size_t)i * HF;
  float a1 = 0.f, a2 = 0.f;
  for (int t = 0; t < HF; ++t) { float v = hr[t]; a1 += v * a[t]; a2 += v * a[HF + t]; }
  s1[i] = a1; s2[i] = a2;
}

__global__ void att_build_k(const float* __restrict__ adj, const float* __restrict__ s1,
                            const float* __restrict__ s2, float* __restrict__ att) {
  size_t e = (size_t)blockIdx.x * blockDim.x + threadIdx.x;
  int i = (int)(e / NV), j = (int)(e % NV);
  float v = 0.0f;
  if (adj[e] > 0.0f) {  // adj symmetric -> symmetric mask
    float e1 = 1.0f / (1.0f + expf(-(s1[i] + s2[j])));
    float e2 = 1.0f / (1.0f + expf(-(s1[j] + s2[i])));
    v = 0.5f * (e1 + e2);
  }
  if (i == j) v = 1.0f;
  att[e] = v;
}

// ---------------- exact percentile via 2x16-bit radix select ----------------
__global__ void zero_scal_k(Scal* s) { s->npos = 0; }

__global__ void count_pos_k(const float* __restrict__ att, int* __restrict__ npos) {
  size_t e = (size_t)blockIdx.x * blockDim.x + threadIdx.x;
  unsigned long long m = __ballot(att[e] > 0.0f);
  if ((threadIdx.x & 31) == 0) atomicAdd(npos, (int)__popcll(m));
}

__global__ void calc_k_k(Scal* s) {
  float idx = (float)(s->npos - 1) * 0.30f;   // P_CUT/100
  int lo = (int)floorf(idx), hi = (int)ceilf(idx);
  s->klo = lo; s->khi = hi; s->frac = idx - (float)lo;
}

__global__ void hist_hi_k(const float* __restrict__ att, unsigned* __restrict__ hist) {
  size_t e = (size_t)blockIdx.x * blockDim.x + threadIdx.x;
  float v = att[e];
  if (v > 0.0f) atomicAdd(&hist[__float_as_uint(v) >> 16], 1u);
}

__global__ void scan_hi_k(const unsigned* __restrict__ hist, Scal* s) {
  long long c = 0;
  const long long kl = s->klo, kh = s->khi;
  for (int b = 0; b < 65536; ++b) {
    long long c2 = c + hist[b];
    if (kl >= c && kl < c2) { s->pref_lo = (unsigned)b; s->rem_lo = (int)(kl - c); }
    if (kh >= c && kh < c2) { s->pref_hi = (unsigned)b; s->rem_hi = (int)(kh - c); }
    c = c2;
  }
}

__global__ void hist_lo_k(const float* __restrict__ att, const Scal* __restrict__ s,
                          unsigned* __restrict__ hl, unsigned* __restrict__ hh) {
  size_t e = (size_t)blockIdx.x * blockDim.x + threadIdx.x;
  float v = att[e];
  if (v <= 0.0f) return;
  unsigned b = __float_as_uint(v), hi16 = b >> 16;
  if (hi16 == s->pref_lo) atomicAdd(&hl[b & 0xFFFFu], 1u);
  if (hi16 == s->pref_hi) atomicAdd(&hh[b & 0xFFFFu], 1u);
}

__global__ void scan_lo_k(const unsigned* __restrict__ hl, const unsigned* __restrict__ hh,
                          Scal* s) {
  long long c = 0; unsigned blo = 0, bhi = 0;
  for (int b = 0; b < 65536; ++b) {
    long long c2 = c + hl[b];
    if (s->rem_lo >= c && s->rem_lo < c2) blo = (unsigned)b;
    c = c2;
  }
  c = 0;
  for (int b = 0; b < 65536; ++b) {
    long long c2 = c + hh[b];
    if (s->rem_hi >= c && s->rem_hi < c2) bhi = (unsigned)b;
    c = c2;
  }
  float vlo = __uint_as_float((s->pref_lo << 16) | blo);
  float vhi = __uint_as_float((s->pref_hi << 16) | bhi);
  s->cut = vlo * (1.0f - s->frac) + vhi * s->frac;
}

__global__ void threshold_k(const float* __restrict__ att, const Scal* __restrict__ s,
                            float* __restrict__ out) {
  size_t e = (size_t)blockIdx.x * blockDim.x + threadIdx.x;
  float v = att[e];
  out[e] = (v >= s->cut) ? v : 0.0f;
}

__global__ __launch_bounds__(256) void row_invcnt_k(const float* __restrict__ ac,
                                                    float* __restrict__ inv) {
  __shared__ int sh[256];
  const int i = blockIdx.x;
  int c = 0;
  for (int j = threadIdx.x; j < NV; j += 256) c += (ac[(size_t)i * NV + j] > 0.0f);
  sh[threadIdx.x] = c;
  __syncthreads();
  for (int s = 128; s > 0; s >>= 1) { if (threadIdx.x < s) sh[threadIdx.x] += sh[threadIdx.x + s]; __syncthreads(); }
  if (threadIdx.x == 0) inv[i] = 1.0f / (float)sh[0];
}

// ---------------- connected components (min-prop + pointer jumping) ---------
__global__ void iota_k(int* __restrict__ L) {
  int i = blockIdx.x * blockDim.x + threadIdx.x;
  if (i < NV) L[i] = i;
}

__global__ __launch_bounds__(256) void cc_min_k(const float* __restrict__ ac,
                                                const int* __restrict__ L,
                                                int* __restrict__ T) {
  __shared__ int sh[256];
  const int i = blockIdx.x;
  int m = L[i];
  for (int j = threadIdx.x; j < NV; j += 256)
    if (ac[(size_t)i * NV + j] > 0.0f) m = min(m, L[j]);
  sh[threadIdx.x] = m;
  __syncthreads();
  for (int s = 128; s > 0; s >>= 1) { if (threadIdx.x < s) sh[threadIdx.x] = min(sh[threadIdx.x], sh[threadIdx.x + s]); __syncthreads(); }
  if (threadIdx.x == 0) T[i] = sh[0];
}

__global__ void cc_jump_k(const int* __restrict__ T, int* __restrict__ L) {
  int i = blockIdx.x * blockDim.x + threadIdx.x;
  if (i < NV) L[i] = T[T[i]];
}

// ---------------- pooling scatter/gather ------------------------------------
__global__ void set_C_k(const int* __restrict__ L, float* __restrict__ Cm) {
  int j = blockIdx.x * blockDim.x + threadIdx.x;
  if (j < NV) Cm[(size_t)L[j] * NV + j] = 1.0f;   // C = one_hot(labels).T
}

__global__ void scatter_rows_k(const int* __restrict__ L, const float* __restrict__ src,
                               float* __restrict__ dst) {
  int idx = blockIdx.x * blockDim.x + threadIdx.x;   // over NV*H
  int i = idx / H, c = idx % H;
  atomicAdd(&dst[(size_t)L[i] * H + c], src[idx]);   // x2 = C @ M
}

__global__ void scatter_a2_k(const int* __restrict__ L, const float* __restrict__ adj,
                             float* __restrict__ a2) {
  size_t e = (size_t)blockIdx.x * blockDim.x + threadIdx.x;
  float v = adj[e];
  if (v != 0.0f) {
    int i = (int)(e / NV), j = (int)(e % NV);
    atomicAdd(&a2[(size_t)L[i] * NV + L[j]], v);     // a2 = C adj C^T
  }
}

__global__ __launch_bounds__(256) void rowsum_D_k(const float* __restrict__ a2,
                                                  float* __restrict__ D) {
  __shared__ float sh[256];
  const int i = blockIdx.x;
  float s = 0.0f;
  for (int j = threadIdx.x; j < NV; j += 256) s += a2[(size_t)i * NV + j];
  sh[threadIdx.x] = s;
  __syncthreads();
  for (int st = 128; st > 0; st >>= 1) { if (threadIdx.x < st) sh[threadIdx.x] += sh[threadIdx.x + st]; __syncthreads(); }
  if (threadIdx.x == 0) D[i] = sqrtf(sh[0] + 1.0f);  // +1 from the k*I diagonal
}

__global__ void norm_adj_k(const float* __restrict__ a2, const float* __restrict__ D,
                           float* __restrict__ adjC) {
  size_t e = (size_t)blockIdx.x * blockDim.x + threadIdx.x;
  int i = (int)(e / NV), j = (int)(e % NV);
  float v = a2[e] + ((i == j) ? 1.0f : 0.0f);
  adjC[e] = v / (D[i] * D[j]);
}

__global__ void gather_rows_k(const int* __restrict__ L, const float* __restrict__ src,
                              float* __restrict__ dst) {
  int idx = blockIdx.x * blockDim.x + threadIdx.x;   // over NV*H
  int i = idx / H, c = idx % H;
  dst[idx] = src[(size_t)L[i] * H + c];              // unpool: C^T @ x
}

__global__ void logsoftmax16_k(const float* __restrict__ X, float* __restrict__ out) {
  int i = blockIdx.x * blockDim.x + threadIdx.x;
  if (i >= NV) return;
  const float* x = X + (size_t)i * OD;
  float mx = x[0];
  for (int t = 1; t < OD; ++t) mx = fmaxf(mx, x[t]);
  float s = 0.0f;
  for (int t = 0; t < OD; ++t) s += expf(x[t] - mx);
  float ls = logf(s);
  for (int t = 0; t < OD; ++t) out[(size_t)i * OD + t] = x[t] - mx - ls;
}

// ---------------- host orchestration ----------------------------------------
extern "C" void kernel_launch(void* const* d_in, const int* in_sizes, int n_in,
                              void* d_out, int out_size, void* d_ws, size_t ws_size,
                              hipStream_t stream) {
  (void)in_sizes; (void)n_in; (void)out_size; (void)ws_size;

  const float* xin     = (const float*)d_in[0];
  const float* adjC0   = (const float*)d_in[1];
  const float* adj0    = (const float*)d_in[2];
  const float* encW0   = (const float*)d_in[3];
  const float* encb0   = (const float*)d_in[4];
  const float* encW    = (const float*)d_in[5];
  const float* encb    = (const float*)d_in[6];
  const float* fcW     = (const float*)d_in[7];
  const float* fcb     = (const float*)d_in[8];
  const float* aW      = (const float*)d_in[9];
  const float* decW    = (const float*)d_in[10];
  const float* decb    = (const float*)d_in[11];
  const float* decWo   = (const float*)d_in[12];
  const float* decbo   = (const float*)d_in[13];

  float* out_logp = (float*)d_out;
  float* out_Cs   = out_logp + (size_t)NV * OD;
  float* out_atts = out_Cs + 3 * NN;

  // ---- carve workspace ----
  char* w = (char*)d_ws;
  auto carve = [&](size_t bytes) { void* p = (void*)w; w += (bytes + 255) & ~(size_t)255; return p; };
  float* adjC1 = (float*)carve(NN * 4);
  float* adjC2 = (float*)carve(NN * 4);
  float* adjC3 = (float*)carve(NN * 4);
  float* pingA = (float*)carve(NN * 4);
  float* pingB = (float*)carve(NN * 4);
  float* t0    = (float*)carve((size_t)NV * H * 4);
  float* t1    = (float*)carve((size_t)NV * H * 4);
  float* xwork = (float*)carve((size_t)NV * H * 4);
  float* xs[4];
  for (int k = 0; k < 4; ++k) xs[k] = (float*)carve((size_t)NV * H * 4);
  float* hfc  = (float*)carve((size_t)NV * HF * 4);
  float* s1v  = (float*)carve(NV * 4);
  float* s2v  = (float*)carve(NV * 4);
  float* invc = (float*)carve(NV * 4);
  float* Dv   = (float*)carve(NV * 4);
  int* Lbl    = (int*)carve(NV * 4);
  int* Tmp    = (int*)carve(NV * 4);
  int* labSav[3];
  for (int k = 0; k < 3; ++k) labSav[k] = (int*)carve(NV * 4);
  unsigned* hist  = (unsigned*)carve(65536 * 4);
  unsigned* histL = (unsigned*)carve(65536 * 4);
  unsigned* histH = (unsigned*)carve(65536 * 4);
  Scal* sc = (Scal*)carve(sizeof(Scal));

  const int gNN = (int)(NN / 256);           // 36864
  const int gN  = (NV + 255) / 256;          // 12
  const int gNH = (NV * H) / 256;            // 3072

  auto gemm4 = [&](const float* A, int lda, const float* B, int ldb, float* C, int ldc,
                   int M, int K, int Nc, const float* bias, const float* rsc, int flags) {
    int waves = (M / 16) * (Nc / 64);
    gemm_wmma<4><<<(waves + 7) / 8, 256, 0, stream>>>(A, lda, B, ldb, C, ldc, M, K, Nc, bias, rsc, flags);
  };
  auto gemm1 = [&](const float* A, int lda, const float* B, int ldb, float* C, int ldc,
                   int M, int K, int Nc, const float* bias, int flags) {
    int waves = (M / 16) * (Nc / 16);
    gemm_wmma<1><<<(waves + 7) / 8, 256, 0, stream>>>(A, lda, B, ldb, C, ldc, M, K, Nc, bias, nullptr, flags);
  };
  // big N x N GEMM with async-LDS-staged B (K = NV, Nc multiple of 64)
  auto gemmNN = [&](const float* A, const float* B, float* C, int Nc,
                    const float* bias, const float* rsc, int flags) {
    int blocks = (NV / 128) * (Nc / 64);
    gemm_nn_async<<<blocks, 256, 0, stream>>>(A, B, C, Nc, bias, rsc, flags);
  };

  const float* adjC_list[4] = {adjC0, adjC1, adjC2, adjC3};
  const float* adjCur_list[3] = {adj0, pingA, pingB};
  float* scratch_list[3] = {pingA, pingB, pingA};   // attcut, then next-level adj
  float* adjCn_list[3] = {adjC1, adjC2, adjC3};

  // ================= encode =================
  for (int k = 0; k < 3; ++k) {
    // h = relu(adjC @ (x @ W) + b)
    if (k == 0) gemm4(xin, FIN, encW0, H, t0, H, NV, FIN, H, nullptr, nullptr, 0);
    else        gemm4(xwork, H, encW + (size_t)(k - 1) * H * H, H, t0, H, NV, H, H, nullptr, nullptr, 0);
    gemmNN(adjC_list[k], t0, t1, H, (k == 0 ? encb0 : encb + (size_t)(k - 1) * H), nullptr, GF_RELU);
    // x = l2norm(h [+ x_prev]); xs[k] = x
    l2norm_k<<<NV, 256, 0, stream>>>(t1, (k == 0 ? nullptr : xwork), xs[k]);

    const float* adjCur = adjCur_list[k];
    float* scratch = scratch_list[k];        // holds attcut first, adj_{k+1} later
    float* attOut = out_atts + (size_t)k * NN;
    float* COut   = out_Cs + (size_t)k * NN;

    // ---- edge_score ----
    gemm4(xs[k], H, fcW + (size_t)k * H * HF, HF, hfc, HF, NV, H, HF, fcb + (size_t)k * HF, nullptr, 0);
    edge_s_k<<<gN, 256, 0, stream>>>(hfc, aW + (size_t)k * 2 * HF, s1v, s2v);
    att_build_k<<<gNN, 256, 0, stream>>>(adjCur, s1v, s2v, attOut);

    // ---- percentile (exact radix select over positive entries) ----
    zero_scal_k<<<1, 1, 0, stream>>>(sc);
    count_pos_k<<<gNN, 256, 0, stream>>>(attOut, &sc->npos);
    calc_k_k<<<1, 1, 0, stream>>>(sc);
    fill_u_k<<<256, 256, 0, stream>>>(hist, 65536);
    hist_hi_k<<<gNN, 256, 0, stream>>>(attOut, hist);
    scan_hi_k<<<1, 1, 0, stream>>>(hist, sc);
    fill_u_k<<<256, 256, 0, stream>>>(histL, 65536);
    fill_u_k<<<256, 256, 0, stream>>>(histH, 65536);
    hist_lo_k<<<gNN, 256, 0, stream>>>(attOut, sc, histL, histH);
    scan_lo_k<<<1, 1, 0, stream>>>(histL, histH, sc);

    // ---- threshold, counts, connected components ----
    threshold_k<<<gNN, 256, 0, stream>>>(attOut, sc, scratch);
    row_invcnt_k<<<NV, 256, 0, stream>>>(scratch, invc);
    iota_k<<<gN, 256, 0, stream>>>(Lbl);
    for (int it = 0; it < 16; ++it) {
      cc_min_k<<<NV, 256, 0, stream>>>(scratch, Lbl, Tmp);
      cc_jump_k<<<gN, 256, 0, stream>>>(Tmp, Lbl);
    }
    (void)hipMemcpyAsync(labSav[k], Lbl, NV * sizeof(int), hipMemcpyDeviceToDevice, stream);
    fill_f_k<<<gNN, 256, 0, stream>>>(COut, 0.0f, NN);
    set_C_k<<<gN, 256, 0, stream>>>(Lbl, COut);

    // ---- x2 = C @ ((att/cnt) @ x) : WMMA GEMM w/ fused row scale, then scatter
    gemmNN(scratch, xs[k], t0, H, nullptr, invc, 0);
    fill_f_k<<<gNH, 256, 0, stream>>>(t1, 0.0f, (size_t)NV * H);
    scatter_rows_k<<<gNH, 256, 0, stream>>>(Lbl, t0, t1);

    // ---- adj_{k+1} = C adj C^T (scatter) + symmetric normalize ----
    fill_f_k<<<gNN, 256, 0, stream>>>(scratch, 0.0f, NN);          // attcut dead now
    scatter_a2_k<<<gNN, 256, 0, stream>>>(Lbl, adjCur, scratch);
    rowsum_D_k<<<NV, 256, 0, stream>>>(scratch, Dv);
    norm_adj_k<<<gNN, 256, 0, stream>>>(scratch, Dv, adjCn_list[k]);

    l2norm_k<<<NV, 256, 0, stream>>>(t1, nullptr, xwork);          // x = l2norm(x2)
  }

  // ================= bottleneck =================
  gemm4(xwork, H, encW + (size_t)2 * H * H, H, t0, H, NV, H, H, nullptr, nullptr, 0);
  gemmNN(adjC3, t0, t1, H, encb + (size_t)2 * H, nullptr, GF_ELU);
  l2norm_k<<<NV, 256, 0, stream>>>(t1, nullptr, xs[3]);

  // ================= decode =================
  const float* xdec = xs[3];
  for (int k = 0; k < 3; ++k) {
    const int lev = 2 - k;
    gather_rows_k<<<gNH, 256, 0, stream>>>(labSav[lev], xdec, t1);   // xu = C^T x
    const float* Wk = decW + (size_t)k * 2 * H * H;
    gemm4(t1, H, Wk, H, t0, H, NV, H, H, nullptr, nullptr, 0);               // xu @ W_top
    gemm4(xs[lev], H, Wk + (size_t)H * H, H, t0, H, NV, H, H, nullptr, nullptr, GF_ACC); // + xs @ W_bot
    gemmNN(adjC_list[lev], t0, t1, H, decb + (size_t)k * H, nullptr, GF_ELU);
    l2norm_k<<<NV, 256, 0, stream>>>(t1, nullptr, xwork);
    xdec = xwork;
  }

  // ================= output head =================
  gemm1(xdec, H, decWo, OD, t0, OD, NV, H, OD, nullptr, 0);
  gemm1(adjC0, NV, t0, OD, t1, OD, NV, NV, OD, decbo, 0);
  logsoftmax16_k<<<gN, 256, 0, stream>>>(t1, out_logp);
}